// SuperPointMatchesGenerator_58067957842194
// MI455X (gfx1250) — compile-verified
//
#include <hip/hip_runtime.h>
#include <hip/hip_bf16.h>

#define NPTS  2048
#define BATCH 32
#define BN    (NPTS * BATCH)

typedef __attribute__((ext_vector_type(2))) float v2f;
typedef __attribute__((ext_vector_type(8))) float v8f;

// ---------------------------------------------------------------------------
// Kernel 1: batched nearest-neighbor (min + argmin over 2048 columns).
// V_WMMA_F32_16X16X4_F32 computes the FULL squared distance in one op by
// packing the norm terms into the unused K=2,3 slots:
//   A(row) = (-2x, -2y, 1, |a|^2),  B(col) = (bx, by, |b|^2, 1)
//   C[row,col] = -2 a.b + |b|^2 + |a|^2 = d^2
// Each wave owns TWO 16-row tiles (32 rows): 2 WMMAs share one B operand,
// halving the ds/bsq/select overhead per output element.
//   dir 0: rows = kpts0_transformed, cols = kpts1  -> nn0 / min_dist0
//   dir 1: rows = kpts1_transformed, cols = kpts0  -> nn1 / min_dist1
// Block = 256 threads = 8 waves = 16 row tiles; grid = (8, 32, 2).
// ---------------------------------------------------------------------------
__global__ __launch_bounds__(256) void nn_argmin_wmma(
    const float2* __restrict__ k0,  const float2* __restrict__ k1,
    const float2* __restrict__ k0t, const float2* __restrict__ k1t,
    int*   __restrict__ nn0,   int*   __restrict__ nn1,
    float* __restrict__ dist0, float* __restrict__ dist1)
{
    __shared__ float2 sB[NPTS];   // 16 KB of the WGP's 320 KB LDS

    const int b   = blockIdx.y;
    const int dir = blockIdx.z;

    const float2* Apts = (dir == 0 ? k0t : k1t) + b * NPTS;
    const float2* Bpts = (dir == 0 ? k1  : k0 ) + b * NPTS;
    int*   nn   = (dir == 0 ? nn0   : nn1  ) + b * NPTS;
    float* dist = (dir == 0 ? dist0 : dist1) + b * NPTS;

    const int wave = threadIdx.x >> 5;
    const int lane = threadIdx.x & 31;
    const int m    = lane & 15;    // row (A) / col (B) within a tile
    const int h    = lane >> 4;    // half-wave: 0 -> K0,K1  1 -> K2,K3
    const int rowBase = (blockIdx.x * 16 + wave * 2) * 16;   // 32 rows/wave

    // Warm the cache for this wave's A rows while we stage B into LDS.
    __builtin_prefetch(&Apts[rowBase + m], 0, 0);
    __builtin_prefetch(&Apts[rowBase + 16 + m], 0, 0);

    // Stage all 2048 column points into LDS once per block.
    #pragma unroll
    for (int k = 0; k < NPTS / 256; ++k)
        sB[threadIdx.x + 256 * k] = Bpts[threadIdx.x + 256 * k];
    __syncthreads();

    // A operands for the two row tiles (K-packed norm terms in upper half).
    const float2 a0 = Apts[rowBase + m];
    const float2 a1 = Apts[rowBase + 16 + m];
    const float asq0 = a0.x * a0.x + a0.y * a0.y;
    const float asq1 = a1.x * a1.x + a1.y * a1.y;
    v2f A0, A1;
    A0.x = (h == 0) ? (-2.0f * a0.x) : 1.0f;
    A0.y = (h == 0) ? (-2.0f * a0.y) : asq0;
    A1.x = (h == 0) ? (-2.0f * a1.x) : 1.0f;
    A1.y = (h == 0) ? (-2.0f * a1.y) : asq1;

    float minv[16];     // [0..7] tile0 rows, [8..15] tile1 rows
    int   mini[16];
    #pragma unroll
    for (int v = 0; v < 16; ++v) { minv[v] = 3.402823466e38f; mini[v] = 0; }

    // Software-pipelined column-tile loop (prefetch next tile's point).
    float2 bpt = sB[m];
    #pragma unroll 2
    for (int t = 0; t < NPTS / 16; ++t) {
        const float2 cur = bpt;
        const int col = t * 16 + m;
        bpt = sB[(col + 16) & (NPTS - 1)];           // next tile (wraps, harmless)

        const float bsq = cur.x * cur.x + cur.y * cur.y;
        v2f Bop;
        Bop.x = (h == 0) ? cur.x : bsq;
        Bop.y = (h == 0) ? cur.y : 1.0f;

        v8f c0 = {};
        v8f c1 = {};
        c0 = __builtin_amdgcn_wmma_f32_16x16x4_f32(
                false, A0, false, Bop, (short)0, c0, false, false);
        c1 = __builtin_amdgcn_wmma_f32_16x16x4_f32(
                false, A1, false, Bop, (short)0, c1, false, false);

        // C element (vgpr v, lane l): row = v + 8*h, col = l&15 of this tile.
        // Strict '<' + increasing t keeps the first index on ties (jnp.argmin).
        #pragma unroll
        for (int v = 0; v < 8; ++v) {
            const float d0 = c0[v];
            const float d1 = c1[v];
            if (d0 < minv[v])     { minv[v]     = d0; mini[v]     = col; }
            if (d1 < minv[v + 8]) { minv[v + 8] = d1; mini[v + 8] = col; }
        }
    }

    // Reduce across the 16 column lanes of each half-wave (xor masks < 16
    // stay within a half).  Prefer the smaller index on exact ties.
    #pragma unroll
    for (int u = 0; u < 16; ++u) {
        #pragma unroll
        for (int off = 1; off < 16; off <<= 1) {
            const float ov = __shfl_xor(minv[u], off, 32);
            const int   oi = __shfl_xor(mini[u], off, 32);
            if (ov < minv[u] || (ov == minv[u] && oi < mini[u])) {
                minv[u] = ov; mini[u] = oi;
            }
        }
    }

    if (m == 0) {   // lanes 0 and 16 hold the reduced results
        #pragma unroll
        for (int v = 0; v < 8; ++v) {
            const int r0 = rowBase + 8 * h + v;          // tile 0
            dist[r0] = sqrtf(fmaxf(minv[v], 0.0f));
            nn[r0]   = mini[v];
            const int r1 = rowBase + 16 + 8 * h + v;     // tile 1
            dist[r1] = sqrtf(fmaxf(minv[v + 8], 0.0f));
            nn[r1]   = mini[v + 8];
        }
    }
}

// ---------------------------------------------------------------------------
// Kernel 2: mutual-NN cross check + mask logic, faithful to the reference's
// statement order (cc1 uses gt_matches0 AFTER the UNMATCHED update but
// BEFORE the mask updates).
// ---------------------------------------------------------------------------
__global__ __launch_bounds__(256) void crosscheck(
    const int* __restrict__ nn0, const int* __restrict__ nn1,
    const int* __restrict__ mask0, const int* __restrict__ mask1,
    float* __restrict__ g0, float* __restrict__ g1)
{
    const int idx = blockIdx.x * blockDim.x + threadIdx.x;
    if (idx >= BN) return;
    const int b = idx >> 11;          // / 2048
    const int i = idx & (NPTS - 1);

    const int* n0 = nn0 + b * NPTS;
    const int* n1 = nn1 + b * NPTS;
    const int* m0 = mask0 + b * NPTS;
    const int* m1 = mask1 + b * NPTS;

    // gt_matches0[i]
    const int mm = n0[i];
    int g = (n1[mm] == i) ? mm : -1;
    if (!m0[i])  g = -2;
    if (!m1[mm]) g = -2;
    g0[idx] = (float)g;

    // gt_matches1[i]: cc1 = (i == gt_matches0_preMask[n1[i]])
    const int k   = n1[i];
    const int mk  = n0[k];
    const int g0k = (n1[mk] == k) ? mk : -1;   // gt_matches0 pre-mask at k
    int gg = (g0k == i) ? k : -1;
    if (!m1[i]) gg = -2;
    if (!m0[k]) gg = -2;
    g1[idx] = (float)gg;
}

// ---------------------------------------------------------------------------
// Launch.  Inputs (setup_inputs order):
//   0: kpts0 [32,2048,2] f32      1: kpts1 [32,2048,2] f32
//   2: kpts0_transformed f32      3: kpts1_transformed f32
//   4: mask0 [32,2048] int        5: mask1 [32,2048] int
// Output (floats, concatenated): gt0 | gt1 | min_dist0 | min_dist1, each B*N.
// Workspace: nn0, nn1 int arrays (2 * B*N * 4 = 512 KB).
// ---------------------------------------------------------------------------
extern "C" void kernel_launch(void* const* d_in, const int* in_sizes, int n_in,
                              void* d_out, int out_size, void* d_ws, size_t ws_size,
                              hipStream_t stream)
{
    const float2* k0  = (const float2*)d_in[0];
    const float2* k1  = (const float2*)d_in[1];
    const float2* k0t = (const float2*)d_in[2];
    const float2* k1t = (const float2*)d_in[3];
    const int* mask0  = (const int*)d_in[4];
    const int* mask1  = (const int*)d_in[5];

    float* out   = (float*)d_out;
    float* g0    = out;            // [B*N]
    float* g1    = out + BN;       // [B*N]
    float* dist0 = out + 2 * BN;   // [B*N]
    float* dist1 = out + 3 * BN;   // [B*N]

    int* nn0 = (int*)d_ws;
    int* nn1 = nn0 + BN;

    dim3 grid1(NPTS / (16 * 16), BATCH, 2);   // (8, 32, 2)
    nn_argmin_wmma<<<grid1, 256, 0, stream>>>(k0, k1, k0t, k1t,
                                              nn0, nn1, dist0, dist1);

    crosscheck<<<(BN + 255) / 256, 256, 0, stream>>>(nn0, nn1, mask0, mask1,
                                                     g0, g1);
}